// ChebConv_5179730559345
// MI455X (gfx1250) — compile-verified
//
#include <hip/hip_runtime.h>

// ---------------- problem constants (from reference) ----------------
#define N_NODES 50000
#define N_EDGES 800000
#define D_FEAT  128            // D_IN == D_OUT == 128
#define K_CHEB  3              // terms T0..T3, h = [N, 512]

typedef __attribute__((ext_vector_type(2))) float v2f;
typedef __attribute__((ext_vector_type(8))) float v8f;

// ---------------- CSR build kernels ----------------
__global__ void zero_i32(int* __restrict__ p, int n) {
    int i = blockIdx.x * blockDim.x + threadIdx.x;
    if (i < n) p[i] = 0;
}

__global__ void count_deg_i(const int* __restrict__ dst, int* __restrict__ cnt, int nEdges) {
    int e = blockIdx.x * blockDim.x + threadIdx.x;
    if (e < nEdges) atomicAdd(&cnt[dst[e]], 1);
}

// single-block exclusive scan: rowptr[0..n] from cnt[0..n-1]
__global__ __launch_bounds__(256) void scan_excl(const int* __restrict__ cnt,
                                                 int* __restrict__ rowptr, int n) {
    __shared__ int buf[256];
    __shared__ int carry;
    if (threadIdx.x == 0) carry = 0;
    __syncthreads();
    for (int base = 0; base < n; base += 256) {
        int i = base + (int)threadIdx.x;
        int v = (i < n) ? cnt[i] : 0;
        buf[threadIdx.x] = v;
        __syncthreads();
        // Hillis-Steele inclusive scan over 256
        for (int off = 1; off < 256; off <<= 1) {
            int t = (threadIdx.x >= (unsigned)off) ? buf[threadIdx.x - off] : 0;
            __syncthreads();
            buf[threadIdx.x] += t;
            __syncthreads();
        }
        int incl = buf[threadIdx.x];
        int excl = incl - v;
        if (i < n) rowptr[i] = carry + excl;
        __syncthreads();                       // all carry reads done
        if (threadIdx.x == 255) carry += incl; // chunk total
        __syncthreads();
    }
    if (threadIdx.x == 0) rowptr[n] = carry;   // == nEdges
}

__global__ void copy_i32(const int* __restrict__ s, int* __restrict__ d, int n) {
    int i = blockIdx.x * blockDim.x + threadIdx.x;
    if (i < n) d[i] = s[i];
}

// esrc[slot] = src[e], slot allocated from per-dst cursor
__global__ void fill_edges(const int* __restrict__ src, const int* __restrict__ dst,
                           int* __restrict__ cursor, int* __restrict__ esrc, int nEdges) {
    int e = blockIdx.x * blockDim.x + threadIdx.x;
    if (e >= nEdges) return;
    int p = atomicAdd(&cursor[dst[e]], 1);
    esrc[p] = src[e];
}

// ---------------- fused gather propagation ----------------
// One wave per destination node (avg deg = 16). Each lane owns a float4 slice of
// the 128-wide feature row; accumulation stays in registers -> zero f32 atomics.
// Fuses the mean scale and the Chebyshev recurrence: out = hasPrev ? 2*mean - prevprev : mean.
__global__ __launch_bounds__(256) void gather_prop(const float* __restrict__ h,
                                                   const int* __restrict__ esrc,
                                                   const int* __restrict__ rowptr,
                                                   const float* __restrict__ prevprev, // may be null
                                                   float* __restrict__ outp, int nNodes) {
    int v    = blockIdx.x * (blockDim.x >> 5) + (threadIdx.x >> 5);
    int lane = threadIdx.x & 31;
    if (v >= nNodes) return;
    int beg = rowptr[v];
    int end = rowptr[v + 1];
    float4 acc = make_float4(0.f, 0.f, 0.f, 0.f);
    for (int j = beg; j < end; ++j) {
        int s = esrc[j];                                   // wave-uniform scalar load
        if (j + 1 < end)                                    // prefetch next source row slice
            __builtin_prefetch(h + (size_t)esrc[j + 1] * D_FEAT + lane * 4, 0, 0);
        const float4 t = *(const float4*)(h + (size_t)s * D_FEAT + lane * 4);
        acc.x += t.x; acc.y += t.y; acc.z += t.z; acc.w += t.w;
    }
    int deg = end - beg;
    float scale = 1.0f / (float)(deg > 1 ? deg : 1);
    float4 r;
    if (prevprev != nullptr) {
        const float4 p = *(const float4*)(prevprev + (size_t)v * D_FEAT + lane * 4);
        r.x = 2.0f * acc.x * scale - p.x;
        r.y = 2.0f * acc.y * scale - p.y;
        r.z = 2.0f * acc.z * scale - p.z;
        r.w = 2.0f * acc.w * scale - p.w;
    } else {
        r.x = acc.x * scale;
        r.y = acc.y * scale;
        r.z = acc.z * scale;
        r.w = acc.w * scale;
    }
    *(float4*)(outp + (size_t)v * D_FEAT + lane * 4) = r;
}

// ---------------- WMMA GEMM: out[N,128] = [T0|T1|T2|T3] @ W^T + b ----------------
// Block = 256 threads = 8 waves. blockIdx.x = row tile (16 nodes), wave = column tile (16 cols).
// V_WMMA_F32_16X16X4_F32: full f32 precision (GEMM is memory-bound; no reason to quantize).
__global__ __launch_bounds__(256) void cheb_gemm(const float* __restrict__ T0,
                                                 const float* __restrict__ T1,
                                                 const float* __restrict__ T2,
                                                 const float* __restrict__ T3,
                                                 const float* __restrict__ W,    // [128, 512] row-major
                                                 const float* __restrict__ bias, // [128]
                                                 float* __restrict__ out) {      // [50000, 128]
    const int lane = threadIdx.x & 31;
    const int wave = threadIdx.x >> 5;          // 0..7 -> column tile
    const int rowTile = blockIdx.x;             // 0..3124 (50000 = 16*3125)
    const int m   = lane & 15;
    const int hi  = lane >> 4;                  // 0: K {k0,k0+1}, M 0..7 ; 1: K {k0+2,k0+3}, M 8..15
    const int row = rowTile * 16 + m;           // A-matrix row for this lane
    const int col = wave * 16 + m;              // B/C/D column j for this lane

    v8f c = {};
    const float* terms[4] = {T0, T1, T2, T3};

#pragma unroll
    for (int t = 0; t < 4; ++t) {
        const float* __restrict__ h    = terms[t];
        const float* __restrict__ wrow = W + (size_t)col * (D_FEAT * (K_CHEB + 1)) + t * D_FEAT;
#pragma unroll 8
        for (int k0 = 0; k0 < D_FEAT; k0 += 4) {
            const int kk = k0 + (hi << 1);
            v2f a  = *(const v2f*)(h + (size_t)row * D_FEAT + kk);   // A 16x4 f32 fragment
            v2f bm = *(const v2f*)(wrow + kk);                       // B 4x16 f32 fragment
            c = __builtin_amdgcn_wmma_f32_16x16x4_f32(
                    /*neg_a=*/false, a, /*neg_b=*/false, bm,
                    /*c_mod=*/(short)0, c, /*reuse_a=*/false, /*reuse_b=*/false);
        }
    }

    const float bj = bias[col];
    // C/D layout: VGPR i -> M = i + 8*hi, N = lane&15
#pragma unroll
    for (int i = 0; i < 8; ++i) {
        const int mr = rowTile * 16 + i + hi * 8;
        out[(size_t)mr * D_FEAT + col] = c[i] + bj;
    }
}

// ---------------- host-side launch ----------------
extern "C" void kernel_launch(void* const* d_in, const int* in_sizes, int n_in,
                              void* d_out, int out_size, void* d_ws, size_t ws_size,
                              hipStream_t stream) {
    const float* x  = (const float*)d_in[0];                 // [N, 128]
    const int*   ei = (const int*)d_in[1];                   // [2, E]
    const float* W  = (const float*)d_in[2];                 // [128, 512]
    const float* b  = (const float*)d_in[3];                 // [128]
    float* out = (float*)d_out;

    const int* src = ei;
    const int* dst = ei + N_EDGES;

    const size_t ND = (size_t)N_NODES * D_FEAT;              // 6.4M floats
    float* B1     = (float*)d_ws;                            // T1
    float* B2     = B1 + ND;                                 // T2
    float* B3     = B2 + ND;                                 // T3
    int*   rowptr = (int*)(B3 + ND);                         // N+1 ints
    int*   cursor = rowptr + (N_NODES + 1);                  // N ints (counts, then fill cursors)
    int*   esrc   = cursor + N_NODES;                        // E ints

    dim3 blk(256);
    dim3 gEdge((N_EDGES + 255) / 256);
    dim3 gNode((N_NODES + 255) / 256);
    dim3 gWaveNode((N_NODES + 7) / 8);                       // 8 waves/block, wave per node

    // ---- build CSR by destination (once per call) ----
    zero_i32<<<gNode, blk, 0, stream>>>(cursor, N_NODES);
    count_deg_i<<<gEdge, blk, 0, stream>>>(dst, cursor, N_EDGES);
    scan_excl<<<1, blk, 0, stream>>>(cursor, rowptr, N_NODES);
    copy_i32<<<gNode, blk, 0, stream>>>(rowptr, cursor, N_NODES);
    fill_edges<<<gEdge, blk, 0, stream>>>(src, dst, cursor, esrc, N_EDGES);

    // ---- fused Chebyshev propagations (pure gathers, no f32 atomics) ----
    // T1 = P x
    gather_prop<<<gWaveNode, blk, 0, stream>>>(x,  esrc, rowptr, nullptr, B1, N_NODES);
    // T2 = 2 P T1 - T0
    gather_prop<<<gWaveNode, blk, 0, stream>>>(B1, esrc, rowptr, x,       B2, N_NODES);
    // T3 = 2 P T2 - T1
    gather_prop<<<gWaveNode, blk, 0, stream>>>(B2, esrc, rowptr, B1,      B3, N_NODES);

    // ---- out = concat(T0..T3) @ W^T + b   (f32 WMMA) ----
    cheb_gemm<<<dim3(N_NODES / 16), blk, 0, stream>>>(x, B1, B2, B3, W, b, out);
}